// VectorQuantizer_70136815944226
// MI455X (gfx1250) — compile-verified
//
#include <hip/hip_runtime.h>
#include <hip/hip_bf16.h>

// ---------- problem constants ----------
#define NTOK   16384      // B*L = 32*512
#define CDIM   64         // E_DIM
#define NE     8192       // N_E
#define LDIM   512
#define BETA   0.25f

// ---------- d_out float offsets (tuple flattened in return order) ----------
#define LOSS_OFF  0
#define ZQ_OFF    1
#define ZQ_N      (32*64*512)                 // 1048576
#define PPL_OFF   (ZQ_OFF + ZQ_N)             // 1048577
#define OH_OFF    (PPL_OFF + 1)               // 1048578
#define OH_N      ((size_t)NTOK * NE)         // 134217728
#define IDX_OFF   (OH_OFF + OH_N)             // 135266306

// ---------- workspace layout (bytes) ----------
#define ZM_BYTES      (NTOK * CDIM * 4)       // 4 MiB
#define NORM_BYTES    (NE * 4)
#define IDX_BYTES     (NTOK * 4)
#define HIST_BYTES    (NE * 4)
#define PART_N        4096
#define PART_BYTES    (PART_N * 4)

typedef __attribute__((ext_vector_type(2))) float v2f;
typedef __attribute__((ext_vector_type(4))) float v4f;
typedef __attribute__((ext_vector_type(8))) float v8f;

// ---- 1. zero accumulators in ws ----
__global__ void vq_zero_kernel(int* hist, int* mask_sum, float* partials) {
    int i = blockIdx.x * blockDim.x + threadIdx.x;
    if (i < NE) hist[i] = 0;
    if (i < PART_N) partials[i] = 0.0f;
    if (i == 0) *mask_sum = 0;
}

// ---- 2. masked transpose: z (B,C,L) -> zm (NTOK x CDIM) ----
__global__ __launch_bounds__(256) void vq_prep_kernel(
    const float* __restrict__ z, const int* __restrict__ mask,
    float* __restrict__ zm) {
    int id = blockIdx.x * blockDim.x + threadIdx.x;   // id = c*NTOK + t
    if (id >= NTOK * CDIM) return;
    int t = id & (NTOK - 1);
    int c = id >> 14;
    int b = t >> 9;
    int l = t & (LDIM - 1);
    float m = (float)mask[t];
    zm[t * CDIM + c] = z[((size_t)b * CDIM + c) * LDIM + l] * m;
}

// ---- 3. per-code squared norms ----
__global__ __launch_bounds__(128) void vq_norms_kernel(
    const float* __restrict__ emb, float* __restrict__ norms) {
    int j = blockIdx.x * blockDim.x + threadIdx.x;
    if (j >= NE) return;
    const v4f* row = (const v4f*)(emb + (size_t)j * CDIM);
    float s = 0.0f;
    #pragma unroll
    for (int k = 0; k < CDIM / 4; ++k) {
        v4f v = row[k];
        s += v.x * v.x + v.y * v.y + v.z * v.z + v.w * v.w;
    }
    norms[j] = s;
}

// ---- 4. WMMA distance GEMM + argmin (wave32, f32 16x16x4) ----
// One wave per PAIR of 16-row tiles (32 tokens): both tiles' A fragments stay
// resident in VGPRs and multiply against the SAME B fragments, doubling
// arithmetic intensity per B load and giving two independent accumulator
// chains so the matrix pipe's accumulate latency is hidden.
// Per 16-code column tile: 2x16 chained v_wmma_f32_16x16x4_f32 produce
// S = zm @ emb^T; score = |e_j|^2 - 2*S; running (min,idx) per C/D slot, then
// 16-lane shuffle-xor min-reduction (lanes 0-15 rows v, lanes 16-31 rows v+8).
__global__ __launch_bounds__(256) void vq_argmin_kernel(
    const float* __restrict__ zm, const float* __restrict__ emb,
    const float* __restrict__ norms, int* __restrict__ idx_out) {
    const int lane = threadIdx.x & 31;
    const int wave = threadIdx.x >> 5;
    const int rowBase0 = (blockIdx.x * 8 + wave) * 32;   // tile pair
    const int rowBase1 = rowBase0 + 16;
    const int half = lane >> 4;     // 0: K 0..1 / rows M=v   1: K 2..3 / rows M=v+8
    const int l15  = lane & 15;

    // A fragments for both row tiles: a*[ks] covers K = ks*4 .. ks*4+3
    v2f a0[16], a1[16];
    {
        const float* ar0 = zm + (size_t)(rowBase0 + l15) * CDIM + half * 2;
        const float* ar1 = zm + (size_t)(rowBase1 + l15) * CDIM + half * 2;
        #pragma unroll
        for (int ks = 0; ks < 16; ++ks) {
            a0[ks].x = ar0[ks * 4 + 0];
            a0[ks].y = ar0[ks * 4 + 1];
            a1[ks].x = ar1[ks * 4 + 0];
            a1[ks].y = ar1[ks * 4 + 1];
        }
    }

    float bestVal0[8], bestVal1[8];
    int   bestIdx0[8], bestIdx1[8];
    #pragma unroll
    for (int v = 0; v < 8; ++v) {
        bestVal0[v] = 3.4e38f; bestIdx0[v] = 0;
        bestVal1[v] = 3.4e38f; bestIdx1[v] = 0;
    }

    for (int jt = 0; jt < NE / 16; ++jt) {
        const int jBase = jt * 16;
        const float* brow = emb + (size_t)(jBase + l15) * CDIM + half * 2;
        // preload all B fragments so loads overlap, then run both WMMA chains
        v2f bf[16];
        #pragma unroll
        for (int ks = 0; ks < 16; ++ks) {
            bf[ks].x = brow[ks * 4 + 0];
            bf[ks].y = brow[ks * 4 + 1];
        }
        const float nj = norms[jBase + l15];

        v8f acc0 = {}, acc1 = {};
        #pragma unroll
        for (int ks = 0; ks < 16; ++ks) {
            acc0 = __builtin_amdgcn_wmma_f32_16x16x4_f32(
                false, a0[ks], false, bf[ks], (short)0, acc0, false, false);
            acc1 = __builtin_amdgcn_wmma_f32_16x16x4_f32(
                false, a1[ks], false, bf[ks], (short)0, acc1, false, false);
        }

        const int j = jBase + l15;
        #pragma unroll
        for (int v = 0; v < 8; ++v) {
            float s0 = nj - 2.0f * acc0[v];
            float s1 = nj - 2.0f * acc1[v];
            if (s0 < bestVal0[v]) { bestVal0[v] = s0; bestIdx0[v] = j; }
            if (s1 < bestVal1[v]) { bestVal1[v] = s1; bestIdx1[v] = j; }
        }
    }

    // reduce across the 16 lanes of each half (xor masks 1,2,4,8 stay in-half)
    #pragma unroll
    for (int v = 0; v < 8; ++v) {
        float bv0 = bestVal0[v], bv1 = bestVal1[v];
        int   bi0 = bestIdx0[v], bi1 = bestIdx1[v];
        #pragma unroll
        for (int m = 8; m >= 1; m >>= 1) {
            float ov0 = __shfl_xor(bv0, m, 32);
            int   oi0 = __shfl_xor(bi0, m, 32);
            float ov1 = __shfl_xor(bv1, m, 32);
            int   oi1 = __shfl_xor(bi1, m, 32);
            if (ov0 < bv0 || (ov0 == bv0 && oi0 < bi0)) { bv0 = ov0; bi0 = oi0; }
            if (ov1 < bv1 || (ov1 == bv1 && oi1 < bi1)) { bv1 = ov1; bi1 = oi1; }
        }
        if (l15 == 0) {
            idx_out[rowBase0 + half * 8 + v] = bi0;
            idx_out[rowBase1 + half * 8 + v] = bi1;
        }
    }
}

// ---- 5. zero-fill one-hot region (537 MB): b128 stores + head/tail peel ----
// OH_OFF % 4 == 2, so peel 2 floats front/back; middle is 16B aligned.
__global__ __launch_bounds__(256) void vq_fill_kernel(float* __restrict__ out) {
    float* base = out + OH_OFF;
    if (blockIdx.x == 0 && threadIdx.x == 0) {
        base[0] = 0.0f; base[1] = 0.0f;
        base[OH_N - 2] = 0.0f; base[OH_N - 1] = 0.0f;
    }
    float4* p = (float4*)(base + 2);
    size_t n4 = (OH_N - 4) / 4;                 // 33554431 float4 stores
    size_t stride = (size_t)gridDim.x * blockDim.x;
    float4 zero4 = make_float4(0.0f, 0.0f, 0.0f, 0.0f);
    for (size_t i = (size_t)blockIdx.x * blockDim.x + threadIdx.x; i < n4; i += stride)
        p[i] = zero4;
}

// ---- 6. scatter: one-hot ones, histogram, mask count, idx output ----
__global__ __launch_bounds__(256) void vq_scatter_kernel(
    const int* __restrict__ idxw, const int* __restrict__ mask,
    int* __restrict__ hist, int* __restrict__ mask_sum, float* __restrict__ out) {
    int t = blockIdx.x * blockDim.x + threadIdx.x;
    if (t >= NTOK) return;
    int i = idxw[t];
    atomicAdd(&hist[i], 1);
    atomicAdd(mask_sum, mask[t]);
    out[(size_t)OH_OFF + (size_t)t * NE + i] = 1.0f;
    out[IDX_OFF + t] = (float)i;
}

// ---- 7. gather z_q (B,C,L) + masked MSE partial sums ----
__global__ __launch_bounds__(256) void vq_gather_kernel(
    const float* __restrict__ z, const int* __restrict__ mask,
    const float* __restrict__ emb, const int* __restrict__ idxw,
    float* __restrict__ out, float* __restrict__ partials) {
    __shared__ float red[256];
    int id = blockIdx.x * blockDim.x + threadIdx.x;   // (b,c,l) flat == z layout
    int b = id >> 15;
    int rem = id & 32767;
    int c = rem >> 9;
    int l = rem & (LDIM - 1);
    int t = b * LDIM + l;
    float e = emb[(size_t)idxw[t] * CDIM + c];
    float zv = z[id];
    out[ZQ_OFF + id] = e;                             // forward of straight-through == z_q
    float d = e - zv;
    red[threadIdx.x] = d * d * (float)mask[t];
    __syncthreads();
    for (int s = 128; s > 0; s >>= 1) {
        if (threadIdx.x < s) red[threadIdx.x] += red[threadIdx.x + s];
        __syncthreads();
    }
    if (threadIdx.x == 0) partials[blockIdx.x] = red[0];
}

// ---- 8. finalize: loss + perplexity ----
__global__ __launch_bounds__(256) void vq_final_kernel(
    const float* __restrict__ partials, const int* __restrict__ hist,
    const int* __restrict__ mask_sum, float* __restrict__ out) {
    __shared__ float red[256];
    // loss partials (fixed-order tree: deterministic)
    float s = 0.0f;
    for (int i = threadIdx.x; i < PART_N; i += 256) s += partials[i];
    red[threadIdx.x] = s;
    __syncthreads();
    for (int k = 128; k > 0; k >>= 1) {
        if (threadIdx.x < k) red[threadIdx.x] += red[threadIdx.x + k];
        __syncthreads();
    }
    float sse = red[0];
    __syncthreads();
    // entropy over code usage
    float h = 0.0f;
    for (int i = threadIdx.x; i < NE; i += 256) {
        float e = (float)hist[i] * (1.0f / (float)NTOK);
        h += e * __logf(e + 1e-10f);
    }
    red[threadIdx.x] = h;
    __syncthreads();
    for (int k = 128; k > 0; k >>= 1) {
        if (threadIdx.x < k) red[threadIdx.x] += red[threadIdx.x + k];
        __syncthreads();
    }
    if (threadIdx.x == 0) {
        float denom = (float)(*mask_sum) * (float)CDIM;
        out[LOSS_OFF] = (1.0f + BETA) * sse / denom;  // emb_loss == commit_loss fwd
        out[PPL_OFF]  = __expf(-red[0]);
    }
}

extern "C" void kernel_launch(void* const* d_in, const int* in_sizes, int n_in,
                              void* d_out, int out_size, void* d_ws, size_t ws_size,
                              hipStream_t stream) {
    const float* z    = (const float*)d_in[0];
    const int*   mask = (const int*)d_in[1];
    const float* emb  = (const float*)d_in[2];
    float* out = (float*)d_out;

    char* ws = (char*)d_ws;
    float* zm      = (float*)ws;
    float* norms   = (float*)(ws + ZM_BYTES);
    int*   idxw    = (int*)(ws + ZM_BYTES + NORM_BYTES);
    int*   hist    = (int*)(ws + ZM_BYTES + NORM_BYTES + IDX_BYTES);
    float* partial = (float*)(ws + ZM_BYTES + NORM_BYTES + IDX_BYTES + HIST_BYTES);
    int*   msum    = (int*)(ws + ZM_BYTES + NORM_BYTES + IDX_BYTES + HIST_BYTES + PART_BYTES);

    vq_zero_kernel<<<(NE + 255) / 256, 256, 0, stream>>>(hist, msum, partial);
    vq_prep_kernel<<<(NTOK * CDIM) / 256, 256, 0, stream>>>(z, mask, zm);
    vq_norms_kernel<<<NE / 128, 128, 0, stream>>>(emb, norms);
    vq_argmin_kernel<<<NTOK / (32 * 8), 256, 0, stream>>>(zm, emb, norms, idxw);
    vq_fill_kernel<<<4096, 256, 0, stream>>>(out);
    vq_scatter_kernel<<<NTOK / 256, 256, 0, stream>>>(idxw, mask, hist, msum, out);
    vq_gather_kernel<<<PART_N, 256, 0, stream>>>(z, mask, emb, idxw, out, partial);
    vq_final_kernel<<<1, 256, 0, stream>>>(partial, hist, msum, out);
}